// GCN_22926535426197
// MI455X (gfx1250) — compile-verified
//
#include <hip/hip_runtime.h>
#include <math.h>

#define N_NODES 100000
#define N_EDGES 1600000
#define F_INDIM 128
#define H_DIM   64
#define C_DIM   40
#define C_PAD   48

typedef __attribute__((ext_vector_type(2))) float v2f;
typedef __attribute__((ext_vector_type(8))) float v8f;

// ---------------- degree / norm ----------------
__global__ void k_deg(const long long* __restrict__ dst, float* __restrict__ deg) {
    int e = blockIdx.x * blockDim.x + threadIdx.x;
    if (e < N_EDGES) atomicAdd(&deg[(int)dst[e]], 1.0f);
}

__global__ void k_dinv(float* __restrict__ deg) {
    int i = blockIdx.x * blockDim.x + threadIdx.x;
    if (i < N_NODES) {
        float d = deg[i] + 1.0f;           // + self loop, always > 0
        deg[i] = rsqrtf(d);                // overwrite in place with D^-1/2
    }
}

// ---------------- GEMM1: h = X[100000,128] @ W1[128,64]  (fp32 WMMA) ----------------
__global__ void __launch_bounds__(128) k_gemm1(const float* __restrict__ X,
                                               const float* __restrict__ W,
                                               float* __restrict__ Hout) {
    __shared__ float xs[16 * F_INDIM];        // 8 KB
    __shared__ float ws[F_INDIM * H_DIM];     // 32 KB
    const int t = threadIdx.x;
    const int row0 = blockIdx.x * 16;         // 100000 / 16 = 6250 exact
    for (int i = t * 4; i < 16 * F_INDIM; i += 128 * 4)
        *(float4*)&xs[i] = *(const float4*)&X[row0 * F_INDIM + i];
    for (int i = t * 4; i < F_INDIM * H_DIM; i += 128 * 4)
        *(float4*)&ws[i] = *(const float4*)&W[i];
    __syncthreads();

    const int lane = t & 31;
    const int wave = t >> 5;
    const int n0 = wave * 16;                 // 4 waves cover N-tiles 0..63
    const int m  = lane & 15;
    const int kb = (lane >> 4) * 2;           // lanes 0-15: K {0,1}; 16-31: K {2,3}
    v8f acc = {};
    for (int k = 0; k < F_INDIM; k += 4) {
        v2f a, b;
        a.x = xs[m * F_INDIM + k + kb];
        a.y = xs[m * F_INDIM + k + kb + 1];
        b.x = ws[(k + kb) * H_DIM + n0 + m];
        b.y = ws[(k + kb + 1) * H_DIM + n0 + m];
        acc = __builtin_amdgcn_wmma_f32_16x16x4_f32(false, a, false, b,
                                                    (short)0, acc, false, false);
    }
    const int mr = (lane >> 4) * 8;           // C/D: VGPR r -> M = r (+8 for hi lanes)
#pragma unroll
    for (int r = 0; r < 8; ++r)
        Hout[(row0 + mr + r) * H_DIM + n0 + m] = acc[r];
}

// ---------------- layer-1 scatter: wave per edge, 64 feats = 2/lane ----------------
__global__ void __launch_bounds__(256) k_agg1(const long long* __restrict__ src,
                                              const long long* __restrict__ dst,
                                              const float* __restrict__ h,
                                              const float* __restrict__ dinv,
                                              float* __restrict__ out) {
    int e = (blockIdx.x * blockDim.x + threadIdx.x) >> 5;
    if (e >= N_EDGES) return;
    int lane = threadIdx.x & 31;
    __builtin_prefetch(&src[e + 4096], 0, 1);   // stream the edge list ahead (global_prefetch_b8)
    int s = (int)src[e], d = (int)dst[e];
    float norm = dinv[s] * dinv[d];
    float2 v = *(const float2*)&h[s * H_DIM + lane * 2];
    atomicAdd(&out[d * H_DIM + lane * 2],     v.x * norm);
    atomicAdd(&out[d * H_DIM + lane * 2 + 1], v.y * norm);
}

// ---------------- layer-1 epilogue: self-loop + bias + relu ----------------
__global__ void k_fuse1(float* __restrict__ agg, const float* __restrict__ h,
                        const float* __restrict__ dinv, const float* __restrict__ b1) {
    int idx = blockIdx.x * blockDim.x + threadIdx.x;
    if (idx >= N_NODES * H_DIM) return;
    int i = idx >> 6, f = idx & 63;
    float di = dinv[i];
    float v = agg[idx] + h[idx] * di * di + b1[f];
    agg[idx] = v > 0.0f ? v : 0.0f;
}

// ---------------- GEMM2: out2 = out1[100000,64] @ W2[64,40] (pad N->48) ----------------
__global__ void __launch_bounds__(96) k_gemm2(const float* __restrict__ Hin,
                                              const float* __restrict__ W,
                                              float* __restrict__ Out) {
    __shared__ float hs[16 * H_DIM];          // 4 KB
    __shared__ float ws[H_DIM * C_PAD];       // 12 KB, cols 40..47 zero
    const int t = threadIdx.x;
    const int row0 = blockIdx.x * 16;
    for (int i = t; i < 16 * H_DIM; i += 96)
        hs[i] = Hin[row0 * H_DIM + i];
    for (int i = t; i < H_DIM * C_PAD; i += 96) {
        int r = i / C_PAD, c = i % C_PAD;
        ws[i] = (c < C_DIM) ? W[r * C_DIM + c] : 0.0f;
    }
    __syncthreads();

    const int lane = t & 31;
    const int wave = t >> 5;                  // 3 waves -> N-tiles 0,16,32
    const int n0 = wave * 16;
    const int m  = lane & 15;
    const int kb = (lane >> 4) * 2;
    v8f acc = {};
    for (int k = 0; k < H_DIM; k += 4) {
        v2f a, b;
        a.x = hs[m * H_DIM + k + kb];
        a.y = hs[m * H_DIM + k + kb + 1];
        b.x = ws[(k + kb) * C_PAD + n0 + m];
        b.y = ws[(k + kb + 1) * C_PAD + n0 + m];
        acc = __builtin_amdgcn_wmma_f32_16x16x4_f32(false, a, false, b,
                                                    (short)0, acc, false, false);
    }
    const int mr = (lane >> 4) * 8;
    const int col = n0 + m;
    if (col < C_DIM) {
#pragma unroll
        for (int r = 0; r < 8; ++r)
            Out[(row0 + mr + r) * C_DIM + col] = acc[r];
    }
}

// ---------------- layer-2 scatter: wave per edge, 40 feats ----------------
__global__ void __launch_bounds__(256) k_agg2(const long long* __restrict__ src,
                                              const long long* __restrict__ dst,
                                              const float* __restrict__ h,
                                              const float* __restrict__ dinv,
                                              float* __restrict__ out) {
    int e = (blockIdx.x * blockDim.x + threadIdx.x) >> 5;
    if (e >= N_EDGES) return;
    int lane = threadIdx.x & 31;
    int s = (int)src[e], d = (int)dst[e];
    float norm = dinv[s] * dinv[d];
    for (int f = lane; f < C_DIM; f += 32)
        atomicAdd(&out[d * C_DIM + f], h[s * C_DIM + f] * norm);
}

// ---------------- final: self-loop + bias + log_softmax (wave per node) ----------------
__global__ void __launch_bounds__(256) k_final(const float* __restrict__ agg,
                                               const float* __restrict__ h,
                                               const float* __restrict__ dinv,
                                               const float* __restrict__ b2,
                                               float* __restrict__ out) {
    int i = (blockIdx.x * blockDim.x + threadIdx.x) >> 5;
    if (i >= N_NODES) return;
    int lane = threadIdx.x & 31;
    float di = dinv[i], d2 = di * di;

    int c0 = lane;                 // 0..31
    int c1 = 32 + lane;            // 32..39 valid for lane<8
    float a0 = agg[i * C_DIM + c0] + h[i * C_DIM + c0] * d2 + b2[c0];
    float a1 = (lane < 8) ? (agg[i * C_DIM + c1] + h[i * C_DIM + c1] * d2 + b2[c1])
                          : -3.402823466e38f;
    float mx = fmaxf(a0, a1);
    for (int off = 16; off > 0; off >>= 1) mx = fmaxf(mx, __shfl_xor(mx, off, 32));
    float se = expf(a0 - mx) + ((lane < 8) ? expf(a1 - mx) : 0.0f);
    for (int off = 16; off > 0; off >>= 1) se += __shfl_xor(se, off, 32);
    float lse = logf(se);
    out[i * C_DIM + c0] = a0 - mx - lse;
    if (lane < 8) out[i * C_DIM + c1] = a1 - mx - lse;
}

extern "C" void kernel_launch(void* const* d_in, const int* in_sizes, int n_in,
                              void* d_out, int out_size, void* d_ws, size_t ws_size,
                              hipStream_t stream) {
    const float*     x    = (const float*)d_in[0];
    const long long* ei   = (const long long*)d_in[1];   // [2, E] int64
    const float*     W1   = (const float*)d_in[2];
    const float*     b1   = (const float*)d_in[3];
    const float*     W2   = (const float*)d_in[4];
    const float*     b2   = (const float*)d_in[5];
    float*           out  = (float*)d_out;

    const long long* src = ei;
    const long long* dst = ei + N_EDGES;

    float* dinv = (float*)d_ws;                // N
    float* buf1 = dinv + N_NODES;              // N*64  (h, later out2[N*40])
    float* buf2 = buf1 + (size_t)N_NODES * 64; // N*64  (agg1/out1, later agg2[N*40])

    // degrees -> D^-1/2
    hipMemsetAsync(dinv, 0, (size_t)N_NODES * sizeof(float), stream);
    k_deg <<<(N_EDGES + 255) / 256, 256, 0, stream>>>(dst, dinv);
    k_dinv<<<(N_NODES + 255) / 256, 256, 0, stream>>>(dinv);

    // layer 1
    k_gemm1<<<N_NODES / 16, 128, 0, stream>>>(x, W1, buf1);
    hipMemsetAsync(buf2, 0, (size_t)N_NODES * H_DIM * sizeof(float), stream);
    k_agg1 <<<(N_EDGES + 7) / 8, 256, 0, stream>>>(src, dst, buf1, dinv, buf2);
    k_fuse1<<<(N_NODES * H_DIM + 255) / 256, 256, 0, stream>>>(buf2, buf1, dinv, b1);

    // layer 2
    k_gemm2<<<N_NODES / 16, 96, 0, stream>>>(buf2, W2, buf1);
    hipMemsetAsync(buf2, 0, (size_t)N_NODES * C_DIM * sizeof(float), stream);
    k_agg2 <<<(N_EDGES + 7) / 8, 256, 0, stream>>>(src, dst, buf1, dinv, buf2);
    k_final<<<(N_NODES + 7) / 8, 256, 0, stream>>>(buf2, buf1, dinv, b2, out);
}